// TensorProductWithScalarComponents_23424751632776
// MI455X (gfx1250) — compile-verified
//
#include <hip/hip_runtime.h>

typedef __attribute__((ext_vector_type(2))) float v2f;
typedef __attribute__((ext_vector_type(4))) float v4f;
typedef __attribute__((ext_vector_type(8))) float v8f;
typedef int v4i_gcc __attribute__((vector_size(4 * sizeof(int))));

#define GLOBAL_AS __attribute__((address_space(1)))
#define LDS_AS    __attribute__((address_space(3)))

// ---------------------------------------------------------------------------
// CDNA5 async global->LDS copy (ASYNCcnt path).
// Probe-learned prototype: param1 = int4 AS1* (global src), param2 = LDS dst.
// ---------------------------------------------------------------------------
__device__ __forceinline__ void async_copy_b128(const float* g, float* l)
{
#if __has_builtin(__builtin_amdgcn_global_load_async_to_lds_b128)
    __builtin_amdgcn_global_load_async_to_lds_b128(
        (GLOBAL_AS v4i_gcc*)(uintptr_t)g,
        (LDS_AS v4i_gcc*)(unsigned)(uintptr_t)l, 0, 0);
#else
    unsigned loff = (unsigned)(uintptr_t)l;
    asm volatile("global_load_async_to_lds_b128 %0, %1, off"
                 :: "v"(loff), "v"(g) : "memory");
#endif
}

__device__ __forceinline__ void wait_async_le16()
{
#if __has_builtin(__builtin_amdgcn_s_wait_asynccnt)
    __builtin_amdgcn_s_wait_asynccnt(16);
#else
    asm volatile("s_wait_asynccnt 0x10" ::: "memory");
#endif
}
__device__ __forceinline__ void wait_async_0()
{
#if __has_builtin(__builtin_amdgcn_s_wait_asynccnt)
    __builtin_amdgcn_s_wait_asynccnt(0);
#else
    asm volatile("s_wait_asynccnt 0x0" ::: "memory");
#endif
}

// ---------------------------------------------------------------------------
// Setup: Wc[v,u,w2] = c * sum_w1 Wt[u,v,w1] * Wl[w1,w2], stored pre-swizzled
// in WMMA f32 16x16x4 A-operand lane layout, w2t-MAJOR stream order:
//   jt = w2t*MBLK + u   (so the whole weight buffer is one linear stream
//                        consumed in w2t-outer / u-inner loop order)
//   flat idx = ((jt*8 + kk)*32 + lane)*4 + q
//   q0: v=8kk  +(half?2:0)  q1: v=8kk  +(half?3:1)
//   q2: v=8kk+4+(half?2:0)  q3: v=8kk+4+(half?3:1)   (half = lane>=16)
//   A-row (M) = lane&15  ->  w2 = w2t*16 + (lane&15)
// ---------------------------------------------------------------------------
__global__ void build_wc_kernel(const float* __restrict__ Wt,   // (m,64,m)
                                const float* __restrict__ Wl,   // (m,m)
                                float* __restrict__ dst,
                                int m, float c)
{
    int idx = blockIdx.x * blockDim.x + threadIdx.x;
    int total = m * m * 64;
    if (idx >= total) return;
    int q    = idx & 3;
    int lane = (idx >> 2) & 31;
    int kk   = (idx >> 7) & 7;
    int jt   = idx >> 10;
    int u    = jt % m;                      // w2t-major stream
    int w2t  = jt / m;
    int w2   = w2t * 16 + (lane & 15);
    int half = lane >> 4;
    int v    = 8 * kk + 4 * (q >> 1) + ((q & 1) ? (half ? 3 : 1) : (half ? 2 : 0));
    const float* wt = Wt + ((size_t)u * 64 + v) * m;
    float s = 0.0f;
    for (int w1 = 0; w1 < m; ++w1)
        s = fmaf(wt[w1], Wl[(size_t)w1 * m + w2], s);
    dst[idx] = c * s;
}

// ---------------------------------------------------------------------------
// Double-buffered 64KB-chunk stager: 84 chunks cover the 5.25MB stream
// (block0: 64 = 8 w2t x 8, block1: 16 = 4 x 4, block2: 4 = 2 x 2).
// ---------------------------------------------------------------------------
#define NCHUNKS      84
#define CHUNK_FLOATS 16384   // 64KB

struct Ctx {
    const float* wstream;
    float*       lbuf;       // 2 * CHUNK_FLOATS floats in LDS
    int          tid;
    int          next;       // next chunk to issue
    int          cur;        // chunk being consumed
};

__device__ __forceinline__ void issue_chunk(Ctx& c, int ch)
{
    const float* g = c.wstream + (size_t)ch * CHUNK_FLOATS + c.tid * 4;
    float*       l = c.lbuf + (ch & 1) * CHUNK_FLOATS + c.tid * 4;
    #pragma unroll
    for (int i = 0; i < 16; ++i)               // 256 thr * 16B * 16 = 64KB
        async_copy_b128(g + i * 1024, l + i * 1024);
}

__device__ __forceinline__ float* acquire(Ctx& c)
{
    if (c.next < NCHUNKS) {                    // uniform branch
        issue_chunk(c, c.next);
        ++c.next;
        wait_async_le16();                     // newest 16 may be pending;
    } else {                                   // everything older is done
        wait_async_0();
    }
    __syncthreads();
    return c.lbuf + (c.cur & 1) * CHUNK_FLOATS;
}

__device__ __forceinline__ void release(Ctx& c)
{
    __syncthreads();                           // protect buffer before reuse
    ++c.cur;
}

// ---------------------------------------------------------------------------
// Main compute: one wave32 owns TWO 16-sample tiles (A,B).
// D[j,n] = sum_v Wc^T[j,v] * y^T[v,n] via chained v_wmma_f32_16x16x4_f32;
// each 16B/lane LDS weight fetch feeds 4 WMMAs (2 tiles x 2 K-chunks).
// Weight loads for a whole u-iteration are batched into aw[8] so the
// scheduler can clause them and hide LDS latency under the 32 WMMAs.
// ---------------------------------------------------------------------------
template <int MBLK, int T>
__device__ __forceinline__ void process_block(
    Ctx& ctx,
    const float (&ybA0)[16], const float (&ybA1)[16],
    const float (&ybB0)[16], const float (&ybB1)[16],
    const float* __restrict__ xA, const float* __restrict__ xB,
    float* __restrict__ oA, float* __restrict__ oB,
    int lane, bool vA, bool vB)
{
    constexpr int W2T = MBLK / 16;

    #pragma unroll 1
    for (int w2t = 0; w2t < W2T; ++w2t) {
        v8f accA[T], accB[T];
        #pragma unroll
        for (int t = 0; t < T; ++t)
            #pragma unroll
            for (int r = 0; r < 8; ++r) { accA[t][r] = 0.0f; accB[t][r] = 0.0f; }

        #pragma unroll 1
        for (int cu = 0; cu < MBLK / 16; ++cu) {
            float* lb = acquire(ctx);
            const float* ap0 = lb + lane * 4;

            #pragma unroll 2
            for (int uu = 0; uu < 16; ++uu) {
                int u = cu * 16 + uu;
                float xvA[T], xvB[T];
                #pragma unroll
                for (int t = 0; t < T; ++t) {
                    xvA[t] = xA[u * T + t];
                    xvB[t] = xB[u * T + t];
                }
                const float* ap = ap0 + uu * 1024;

                v4f aw[8];                      // batched ds_load_b128 clause
                #pragma unroll
                for (int kk = 0; kk < 8; ++kk)
                    aw[kk] = *(const v4f*)(ap + kk * 128);

                v8f dA, dB;
                #pragma unroll
                for (int r = 0; r < 8; ++r) { dA[r] = 0.0f; dB[r] = 0.0f; }
                #pragma unroll
                for (int kk = 0; kk < 8; ++kk) {
                    v2f alo; alo[0] = aw[kk][0]; alo[1] = aw[kk][1];
                    v2f ahi; ahi[0] = aw[kk][2]; ahi[1] = aw[kk][3];
                    v2f beA; beA[0] = ybA0[2*kk];   beA[1] = ybA1[2*kk];
                    v2f boA; boA[0] = ybA0[2*kk+1]; boA[1] = ybA1[2*kk+1];
                    v2f beB; beB[0] = ybB0[2*kk];   beB[1] = ybB1[2*kk];
                    v2f boB; boB[0] = ybB0[2*kk+1]; boB[1] = ybB1[2*kk+1];
                    dA = __builtin_amdgcn_wmma_f32_16x16x4_f32(false, alo, false, beA,
                                                               (short)0, dA, false, false);
                    dA = __builtin_amdgcn_wmma_f32_16x16x4_f32(false, ahi, false, boA,
                                                               (short)0, dA, false, false);
                    dB = __builtin_amdgcn_wmma_f32_16x16x4_f32(false, alo, false, beB,
                                                               (short)0, dB, false, false);
                    dB = __builtin_amdgcn_wmma_f32_16x16x4_f32(false, ahi, false, boB,
                                                               (short)0, dB, false, false);
                }
                #pragma unroll
                for (int t = 0; t < T; ++t)
                    #pragma unroll
                    for (int r = 0; r < 8; ++r) {
                        accA[t][r] = fmaf(dA[r], xvA[t], accA[t][r]);
                        accB[t][r] = fmaf(dB[r], xvB[t], accB[t][r]);
                    }
            }
            release(ctx);
        }

        if (vA) {
            #pragma unroll
            for (int r = 0; r < 8; ++r)
                #pragma unroll
                for (int t = 0; t < T; ++t)
                    oA[(w2t * 16 + r) * T + t] = accA[t][r];
        }
        if (vB) {
            #pragma unroll
            for (int r = 0; r < 8; ++r)
                #pragma unroll
                for (int t = 0; t < T; ++t)
                    oB[(w2t * 16 + r) * T + t] = accB[t][r];
        }
    }
}

__global__ __launch_bounds__(256, 1)
void tp_main_kernel(const float* __restrict__ x, const float* __restrict__ y,
                    const float* __restrict__ wstream, float* __restrict__ out,
                    int N)
{
    __shared__ float lbuf[2 * CHUNK_FLOATS];   // 128KB double buffer

    const int tid  = threadIdx.x;
    const int lane = tid & 31;
    const int wv   = tid >> 5;
    const int half = lane >> 4;

    // Two sample-tiles per wave. No early return: all waves join barriers
    // and copy slices; invalid tiles are clamped for loads, masked on store.
    int t0 = (blockIdx.x * 8 + wv) * 2;
    int nA = t0 * 16 + (lane & 15);
    int nB = nA + 16;
    bool vA = nA < N;  int nAl = vA ? nA : N - 1;
    bool vB = nB < N;  int nBl = vB ? nB : N - 1;

    // y B-operands (K x N layout), loaded once, reused by all 1344 wtiles:
    const float* yrA = y + (size_t)nAl * 64;
    const float* yrB = y + (size_t)nBl * 64;
    float ybA0[16], ybA1[16], ybB0[16], ybB1[16];
    #pragma unroll
    for (int kc = 0; kc < 16; ++kc) {
        ybA0[kc] = yrA[4 * kc + half];
        ybA1[kc] = yrA[4 * kc + 2 + half];
        ybB0[kc] = yrB[4 * kc + half];
        ybB1[kc] = yrB[4 * kc + 2 + half];
    }

    Ctx ctx;
    ctx.wstream = wstream;
    ctx.lbuf    = lbuf;
    ctx.tid     = tid;
    ctx.next    = 0;
    ctx.cur     = 0;
    issue_chunk(ctx, 0);                       // prime the pipeline
    ctx.next = 1;

    const float* xA = x + (size_t)nAl * 480;
    const float* xB = x + (size_t)nBl * 480;
    float* oAb = out + (size_t)nAl * 480;
    float* oBb = out + (size_t)nBl * 480;

    process_block<128, 1>(ctx, ybA0, ybA1, ybB0, ybB1,
                          xA,       xB,
                          oAb       + half * 8 * 1, oBb       + half * 8 * 1,
                          lane, vA, vB);
    process_block< 64, 3>(ctx, ybA0, ybA1, ybB0, ybB1,
                          xA + 128, xB + 128,
                          oAb + 128 + half * 8 * 3, oBb + 128 + half * 8 * 3,
                          lane, vA, vB);
    process_block< 32, 5>(ctx, ybA0, ybA1, ybB0, ybB1,
                          xA + 320, xB + 320,
                          oAb + 320 + half * 8 * 5, oBb + 320 + half * 8 * 5,
                          lane, vA, vB);
}

// ---------------------------------------------------------------------------
extern "C" void kernel_launch(void* const* d_in, const int* in_sizes, int n_in,
                              void* d_out, int out_size, void* d_ws, size_t ws_size,
                              hipStream_t stream)
{
    const float* x   = (const float*)d_in[0];
    const float* y   = (const float*)d_in[1];
    const float* Wt0 = (const float*)d_in[2];
    const float* Wt1 = (const float*)d_in[3];
    const float* Wt2 = (const float*)d_in[4];
    const float* Wl0 = (const float*)d_in[5];
    const float* Wl1 = (const float*)d_in[6];
    const float* Wl2 = (const float*)d_in[7];

    float* asw0 = (float*)d_ws;                       // 1,048,576 f (4MB)
    float* asw1 = asw0 + 128 * 128 * 64;              //   262,144 f (1MB)
    float* asw2 = asw1 + 64 * 64 * 64;                //    65,536 f (256KB)

    // c = c_tp * c_lin = 1/sqrt(64*m) * 1/sqrt(m) = 1/(8m)
    build_wc_kernel<<<(128 * 128 * 64 + 255) / 256, 256, 0, stream>>>(
        Wt0, Wl0, asw0, 128, 1.0f / 1024.0f);
    build_wc_kernel<<<(64 * 64 * 64 + 255) / 256, 256, 0, stream>>>(
        Wt1, Wl1, asw1, 64, 1.0f / 512.0f);
    build_wc_kernel<<<(32 * 32 * 64 + 255) / 256, 256, 0, stream>>>(
        Wt2, Wl2, asw2, 32, 1.0f / 256.0f);

    int N      = in_sizes[0] / 480;
    int ntiles = (N + 15) / 16;                // 6250 tiles
    int wgs    = (ntiles + 15) / 16;           // 16 tiles per 8-wave WG
    tp_main_kernel<<<wgs, 256, 0, stream>>>(x, y, asw0, (float*)d_out, N);
}